// InterpretableGAT_14826227105950
// MI455X (gfx1250) — compile-verified
//
#include <hip/hip_runtime.h>
#include <hip/hip_bf16.h>
#include <math.h>

#define N_NODES 100000
#define E_EDGES 1600000
#define E_TOT   (E_EDGES + N_NODES)   // self-loops appended
#define G_GRAPHS 128
#define IN_CH 32
#define HIDC 32
#define HEADS 4
#define F1 (HEADS * HIDC)             // 128
#define SLOPE 0.2f

typedef float v2f __attribute__((ext_vector_type(2)));
typedef float v8f __attribute__((ext_vector_type(8)));

// ---- order-preserving float<->uint encoding for atomic segment-max ----
__device__ __forceinline__ unsigned enc_f32(float f) {
    unsigned b = __float_as_uint(f);
    return (b & 0x80000000u) ? ~b : (b | 0x80000000u);
}
__device__ __forceinline__ float dec_f32(unsigned k) {
    unsigned b = (k & 0x80000000u) ? (k & 0x7FFFFFFFu) : ~k;
    return __uint_as_float(b);
}

__global__ void fill_zero_u32(unsigned* __restrict__ p, long n) {
    long i = (long)blockIdx.x * blockDim.x + threadIdx.x;
    if (i < n) p[i] = 0u;
}

// ---- FP32 WMMA GEMM: C[M,NC] = A[M,K] @ B[K,NC], M % 16 == 0, K % 4 == 0,
// one wave per 16x16 output tile; grid = (M/16, NC/16), block = 32 ----
__global__ void wmma_gemm_f32(const float* __restrict__ A,
                              const float* __restrict__ B,
                              float* __restrict__ C, int K, int NC) {
    const int lane  = threadIdx.x;      // 0..31 (wave32)
    const int grp   = lane >> 4;        // lane group 0/1
    const int l16   = lane & 15;
    const long rowB = (long)blockIdx.x * 16;
    const int  colB = blockIdx.y * 16;

    v8f c = {0.f, 0.f, 0.f, 0.f, 0.f, 0.f, 0.f, 0.f};
    for (int k0 = 0; k0 < K; k0 += 4) {
        // A 16x4 frag: lanes 0-15 rows M, VGPR0/1 = K {2*grp, 2*grp+1}
        v2f a, b;
        const int ka = k0 + 2 * grp;
        const float* ap = A + (rowB + l16) * (long)K + ka;
        a.x = ap[0];
        a.y = ap[1];
        // B 4x16 frag: lane = column, K = 2*grp + vgpr
        const float* bp = B + (long)ka * NC + colB + l16;
        b.x = bp[0];
        b.y = bp[NC];
        c = __builtin_amdgcn_wmma_f32_16x16x4_f32(
                false, a, false, b, (short)0, c, false, false);
    }
    // C/D layout: VGPR j holds row M = j + 8*grp, col = lane&15
    #pragma unroll
    for (int j = 0; j < 8; j++) {
        C[(rowB + j + 8 * grp) * (long)NC + colB + l16] = c[j];
    }
}

// ---- per-(node,head) attention coefficients: e_src/e_dst = <h, a> ----
template <int H, int C>
__global__ void node_attn_coeff(const float* __restrict__ h,
                                const float* __restrict__ a_src,
                                const float* __restrict__ a_dst,
                                float* __restrict__ esrc,
                                float* __restrict__ edst) {
    int t = blockIdx.x * blockDim.x + threadIdx.x;  // node*H + head
    if (t >= N_NODES * H) return;
    int n = t / H, hd = t % H;
    const float* hp = h + (size_t)n * (H * C) + hd * C;
    float ss = 0.f, sd = 0.f;
    #pragma unroll
    for (int c = 0; c < C; c++) {
        float v = hp[c];
        ss += v * a_src[hd * C + c];
        sd += v * a_dst[hd * C + c];
    }
    esrc[t] = ss;
    edst[t] = sd;
}

// ---- pass 1: leaky-relu edge scores + atomic segment max over dst ----
template <int H>
__global__ void edge_score_max(const int* __restrict__ src,
                               const int* __restrict__ dst,
                               const float* __restrict__ esrc,
                               const float* __restrict__ edst,
                               float* __restrict__ score,
                               unsigned* __restrict__ maxk) {
    int e = blockIdx.x * blockDim.x + threadIdx.x;
    if (e >= E_TOT) return;
    int s, d;
    if (e < E_EDGES) { s = src[e]; d = dst[e]; }
    else             { s = e - E_EDGES; d = s; }   // self-loop
    #pragma unroll
    for (int h = 0; h < H; h++) {
        float v = esrc[s * H + h] + edst[d * H + h];
        v = (v > 0.f) ? v : SLOPE * v;
        score[(size_t)e * H + h] = v;
        atomicMax(&maxk[d * H + h], enc_f32(v));
    }
}

// ---- pass 2: exp(score - max) in place + atomic segment sum (denominator) ----
template <int H>
__global__ void edge_exp_sum(const int* __restrict__ dst,
                             const unsigned* __restrict__ maxk,
                             float* __restrict__ score,
                             float* __restrict__ den) {
    int e = blockIdx.x * blockDim.x + threadIdx.x;
    if (e >= E_TOT) return;
    int d = (e < E_EDGES) ? dst[e] : e - E_EDGES;
    #pragma unroll
    for (int h = 0; h < H; h++) {
        float ex = __expf(score[(size_t)e * H + h] - dec_f32(maxk[d * H + h]));
        score[(size_t)e * H + h] = ex;
        atomicAdd(&den[d * H + h], ex);
    }
}

// ---- pass 3: weighted scatter-add; one wave per edge, lane = channel ----
template <int H, int C>
__global__ void edge_aggregate(const int* __restrict__ src,
                               const int* __restrict__ dst,
                               const float* __restrict__ h,
                               const float* __restrict__ score,
                               const float* __restrict__ den,
                               float* __restrict__ agg) {
    long t = (long)blockIdx.x * blockDim.x + threadIdx.x;
    long e = t >> 5;
    int lane = (int)(t & 31);
    if (e >= E_TOT) return;
    int s, d;
    if (e < E_EDGES) { s = src[e]; d = dst[e]; }
    else             { s = (int)(e - E_EDGES); d = s; }
    #pragma unroll
    for (int hd = 0; hd < H; hd++) {
        float alpha = score[(size_t)e * H + hd] / den[d * H + hd];
        float v = h[(size_t)s * (H * C) + hd * C + lane] * alpha;
        atomicAdd(&agg[(size_t)d * (H * C) + hd * C + lane], v);
    }
}

// ---- bias + ELU, in place ----
__global__ void bias_elu(float* __restrict__ buf, const float* __restrict__ b,
                         int F, long total) {
    long i = (long)blockIdx.x * blockDim.x + threadIdx.x;
    if (i >= total) return;
    float v = buf[i] + b[i % F];
    buf[i] = (v > 0.f) ? v : (__expf(v) - 1.f);
}

// ---- global mean pool (sum + count); one wave per node, lane = channel ----
__global__ void pool_kernel(const float* __restrict__ act,
                            const int* __restrict__ batch,
                            float* __restrict__ sums, float* __restrict__ cnt) {
    long t = (long)blockIdx.x * blockDim.x + threadIdx.x;
    long n = t >> 5;
    int lane = (int)(t & 31);
    if (n >= N_NODES) return;
    int g = batch[n];
    atomicAdd(&sums[g * HIDC + lane], act[(size_t)n * HIDC + lane]);
    if (lane == 0) atomicAdd(&cnt[g], 1.0f);
}

// ---- final classifier: pooled @ Wc + bc ----
__global__ void classifier_kernel(const float* __restrict__ sums,
                                  const float* __restrict__ cnt,
                                  const float* __restrict__ Wc,
                                  const float* __restrict__ bc,
                                  float* __restrict__ out) {
    int g = threadIdx.x;
    if (g >= G_GRAPHS) return;
    float c = fmaxf(cnt[g], 1.0f);
    float acc = 0.f;
    #pragma unroll
    for (int i = 0; i < HIDC; i++) acc += (sums[g * HIDC + i] / c) * Wc[i];
    out[g] = acc + bc[0];
}

extern "C" void kernel_launch(void* const* d_in, const int* in_sizes, int n_in,
                              void* d_out, int out_size, void* d_ws, size_t ws_size,
                              hipStream_t stream) {
    const float* x      = (const float*)d_in[0];
    const int*   ei     = (const int*)  d_in[1];
    const int*   batch  = (const int*)  d_in[2];
    const float* W1     = (const float*)d_in[3];
    const float* a_src1 = (const float*)d_in[4];
    const float* a_dst1 = (const float*)d_in[5];
    const float* b1     = (const float*)d_in[6];
    const float* W2     = (const float*)d_in[7];
    const float* a_src2 = (const float*)d_in[8];
    const float* a_dst2 = (const float*)d_in[9];
    const float* b2     = (const float*)d_in[10];
    const float* Wc     = (const float*)d_in[11];
    const float* bc     = (const float*)d_in[12];
    float* out = (float*)d_out;
    float* ws  = (float*)d_ws;

    const int* src = ei;             // edge_index[0]
    const int* dst = ei + E_EDGES;   // edge_index[1]

    // ---- workspace layout (floats), ~140MB total; all L2-resident ----
    size_t off_h1   = 0;                                  // N*128
    size_t off_agg1 = off_h1   + (size_t)N_NODES * F1;    // N*128
    size_t off_esrc = off_agg1 + (size_t)N_NODES * F1;    // N*HEADS
    size_t off_edst = off_esrc + (size_t)N_NODES * HEADS;
    size_t off_maxk = off_edst + (size_t)N_NODES * HEADS; // uint
    size_t off_den  = off_maxk + (size_t)N_NODES * HEADS;
    size_t off_edge = off_den  + (size_t)N_NODES * HEADS; // E_TOT*HEADS
    size_t off_pool = off_edge + (size_t)E_TOT * HEADS;   // G*HIDC + G
    float*    h1   = ws + off_h1;
    float*    agg1 = ws + off_agg1;
    float*    esrc = ws + off_esrc;
    float*    edst = ws + off_edst;
    unsigned* maxk = (unsigned*)(ws + off_maxk);
    float*    den  = ws + off_den;
    float*    escr = ws + off_edge;
    float*    pool = ws + off_pool;
    float*    cnt  = pool + (size_t)G_GRAPHS * HIDC;
    // layer-2 reuse of layer-1 space (h1 dead after layer-1 aggregation)
    float*    h2   = ws + off_h1;                          // N*32
    float*    agg2 = ws + off_h1 + (size_t)N_NODES * HIDC; // N*32

    auto zfill = [&](void* p, long n) {
        fill_zero_u32<<<(unsigned)((n + 255) / 256), 256, 0, stream>>>((unsigned*)p, n);
    };

    // ================= layer 1 =================
    zfill(agg1, (long)N_NODES * F1);
    zfill(maxk, (long)N_NODES * HEADS);
    zfill(den,  (long)N_NODES * HEADS);
    zfill(pool, (long)G_GRAPHS * HIDC + G_GRAPHS);

    wmma_gemm_f32<<<dim3(N_NODES / 16, F1 / 16), 32, 0, stream>>>(x, W1, h1, IN_CH, F1);
    node_attn_coeff<HEADS, HIDC><<<(N_NODES * HEADS + 255) / 256, 256, 0, stream>>>(
        h1, a_src1, a_dst1, esrc, edst);
    edge_score_max<HEADS><<<(E_TOT + 255) / 256, 256, 0, stream>>>(
        src, dst, esrc, edst, escr, maxk);
    edge_exp_sum<HEADS><<<(E_TOT + 255) / 256, 256, 0, stream>>>(
        dst, maxk, escr, den);
    edge_aggregate<HEADS, HIDC><<<(unsigned)(((long)E_TOT * 32 + 255) / 256), 256, 0, stream>>>(
        src, dst, h1, escr, den, agg1);
    bias_elu<<<(unsigned)(((long)N_NODES * F1 + 255) / 256), 256, 0, stream>>>(
        agg1, b1, F1, (long)N_NODES * F1);

    // ================= layer 2 (heads=1, mean==identity) =================
    zfill(agg2, (long)N_NODES * HIDC);
    zfill(maxk, (long)N_NODES * 1);
    zfill(den,  (long)N_NODES * 1);

    wmma_gemm_f32<<<dim3(N_NODES / 16, HIDC / 16), 32, 0, stream>>>(agg1, W2, h2, F1, HIDC);
    node_attn_coeff<1, HIDC><<<(N_NODES + 255) / 256, 256, 0, stream>>>(
        h2, a_src2, a_dst2, esrc, edst);
    edge_score_max<1><<<(E_TOT + 255) / 256, 256, 0, stream>>>(
        src, dst, esrc, edst, escr, maxk);
    edge_exp_sum<1><<<(E_TOT + 255) / 256, 256, 0, stream>>>(
        dst, maxk, escr, den);
    edge_aggregate<1, HIDC><<<(unsigned)(((long)E_TOT * 32 + 255) / 256), 256, 0, stream>>>(
        src, dst, h2, escr, den, agg2);
    bias_elu<<<(unsigned)(((long)N_NODES * HIDC + 255) / 256), 256, 0, stream>>>(
        agg2, b2, HIDC, (long)N_NODES * HIDC);

    // ================= pool + classifier =================
    pool_kernel<<<(unsigned)(((long)N_NODES * 32 + 255) / 256), 256, 0, stream>>>(
        agg2, batch, pool, cnt);
    classifier_kernel<<<1, G_GRAPHS, 0, stream>>>(pool, cnt, Wc, bc, out);
}